// BertCRFForTokenClassification_64716567216844
// MI455X (gfx1250) — compile-verified
//
#include <hip/hip_runtime.h>
#include <hip/hip_bf16.h>
#include <cstdint>

// Problem constants from the reference
#define CB 1024   // batch
#define CL 512    // sequence length
#define CT 64     // number of tags
#define LDSP 68   // padded LDS row stride (floats) to spread banks

#define LOG2E 1.44269504088896340736f
#define LN2   0.69314718055994530942f

typedef __attribute__((ext_vector_type(2))) float v2f;
typedef __attribute__((ext_vector_type(8))) float v8f;

__device__ __forceinline__ float fast_exp(float x) {
    return __builtin_amdgcn_exp2f(x * LOG2E);
}
__device__ __forceinline__ float fast_log(float x) {
    return __builtin_amdgcn_logf(x) * LN2;
}

__global__ void crf_zero_out(float* out) { out[0] = 0.0f; }

// One workgroup = 128 threads = 4 waves, handles 16 batch rows.
// Wave w computes output tag-block j in [16w, 16w+16).
__global__ __launch_bounds__(128) void crf_forward_kernel(
    const float* __restrict__ em,        // [B, L, T]
    const float* __restrict__ trans,     // [T, T]
    const float* __restrict__ startT,    // [T]
    const float* __restrict__ endT,      // [T]
    const int*   __restrict__ tags,      // [B, L]
    const unsigned char* __restrict__ msk, // [B, L] (bool)
    float* __restrict__ out)             // scalar
{
    __shared__ float alpha[16][LDSP];
    __shared__ float pbuf[16][LDSP];
    __shared__ float rowm[16];

    const int tid  = threadIdx.x;
    const int wave = tid >> 5;       // 0..3 : j-block
    const int lane = tid & 31;
    const int lhi  = lane >> 4;      // 0/1 (half-wave)
    const int llo  = lane & 15;
    const int row  = tid >> 3;       // 0..15 (for 8-lane-per-row phases)
    const int g    = tid & 7;        // 0..7
    const int b0   = blockIdx.x * 16;

    // ---- Preload B fragments: expT[k, j] = exp(transitions[k][j]) ----
    // V_WMMA_F32_16X16X4_F32 B layout (4x16, K x N):
    //   VGPR t, lane l: K = 4*kc + t + 2*(l>>4), N = jb*16 + (l&15)
    v2f bf[16];
    {
        const int n = wave * 16 + llo;
#pragma unroll
        for (int kc = 0; kc < 16; ++kc) {
            const int k0 = kc * 4 + 2 * lhi;
            bf[kc].x = fast_exp(trans[(k0 + 0) * CT + n]);
            bf[kc].y = fast_exp(trans[(k0 + 1) * CT + n]);
        }
    }

    // ---- alpha0[b, j] = startT[j] + em[b, 0, j] ----
#pragma unroll
    for (int k = 0; k < 8; ++k) {
        const int col = g + 8 * k;
        alpha[row][col] = startT[col] + em[(size_t)(b0 + row) * CL * CT + col];
    }
    __syncthreads();

    // ---- Forward recursion over t = 1..L-1 ----
    for (int t = 1; t < CL; ++t) {
        // Phase 1: per-row max and p = exp(alpha - max)
        float v[8];
        float mx = -3.0e38f;
#pragma unroll
        for (int k = 0; k < 8; ++k) {
            v[k] = alpha[row][g + 8 * k];
            mx = fmaxf(mx, v[k]);
        }
        mx = fmaxf(mx, __shfl_xor(mx, 1, 32));
        mx = fmaxf(mx, __shfl_xor(mx, 2, 32));
        mx = fmaxf(mx, __shfl_xor(mx, 4, 32));
#pragma unroll
        for (int k = 0; k < 8; ++k)
            pbuf[row][g + 8 * k] = fast_exp(v[k] - mx);
        if (g == 0) rowm[row] = mx;
        __syncthreads();

        // Per-wave row-mask ballot: bit r (r=0..15) = mask[b0+r, t]
        // (EXEC reconverges before the WMMA chain below)
        unsigned int rowmask;
        {
            bool mb = (lane < 16) ? (msk[(size_t)(b0 + lane) * CL + t] != 0) : false;
            rowmask = (unsigned int)__ballot(mb);
        }

        // Prefetch next step's emission tile (16 rows x 256B)
        if (t + 1 < CL)
            __builtin_prefetch(&em[((size_t)(b0 + (tid & 15)) * CL + (t + 1)) * CT], 0, 1);

        const int j = wave * 16 + llo;

        // Preload row maxima for this lane's 8 output rows (contiguous LDS)
        float rm[8];
#pragma unroll
        for (int vv = 0; vv < 8; ++vv) rm[vv] = rowm[vv + 8 * lhi];

        // Old alpha values (kept where mask==0) — issue loads early
        float oldv[8];
#pragma unroll
        for (int vv = 0; vv < 8; ++vv) oldv[vv] = alpha[vv + 8 * lhi][j];

        // Emission loads for this step (unconditional, clause together)
        float e[8];
        {
            const float* embase = &em[((size_t)(b0 + 8 * lhi) * CL + t) * CT + j];
#pragma unroll
            for (int vv = 0; vv < 8; ++vv)
                e[vv] = embase[(size_t)vv * CL * CT];
        }

        // Phase 2: Q = P(16x64) x expT(64x64), this wave's 16x16 j-block
        v8f acc = {};
#pragma unroll
        for (int kc = 0; kc < 16; ++kc) {
            // A layout (16x4 f32): lane l: M = l&15, K = 4*kc + t' + 2*(l>>4)
            const float* pr = &pbuf[llo][kc * 4 + 2 * lhi];
            v2f a;
            a.x = pr[0];
            a.y = pr[1];
            acc = __builtin_amdgcn_wmma_f32_16x16x4_f32(
                false, a, false, bf[kc], (short)0, acc, false, false);
        }

        // alpha'[r, j] = blend(mask, rowmax + log(q) + em, old)
        // Arithmetic blend (not a select) so the log path cannot be sunk
        // into per-element EXEC branches; an is always finite since q > 0.
        const unsigned int mbits = rowmask >> (8 * lhi);
#pragma unroll
        for (int vv = 0; vv < 8; ++vv) {
            const int r = vv + 8 * lhi;   // C/D layout: VGPR vv -> M = vv (+8 hi half)
            const float an = rm[vv] + fast_log(acc[vv]) + e[vv];
            const float mf = (float)((mbits >> vv) & 1u);
            alpha[r][j] = fmaf(mf, an - oldv[vv], oldv[vv]);
        }
        __syncthreads();
    }

    // ---- log_z[b] = logsumexp_j(alpha[b,j] + endT[j]) ----
    float w[8];
    float mx2 = -3.0e38f;
#pragma unroll
    for (int k = 0; k < 8; ++k) {
        const int col = g + 8 * k;
        w[k] = alpha[row][col] + endT[col];
        mx2 = fmaxf(mx2, w[k]);
    }
    mx2 = fmaxf(mx2, __shfl_xor(mx2, 1, 32));
    mx2 = fmaxf(mx2, __shfl_xor(mx2, 2, 32));
    mx2 = fmaxf(mx2, __shfl_xor(mx2, 4, 32));
    float s = 0.0f;
#pragma unroll
    for (int k = 0; k < 8; ++k) s += fast_exp(w[k] - mx2);
    s += __shfl_xor(s, 1, 32);
    s += __shfl_xor(s, 2, 32);
    s += __shfl_xor(s, 4, 32);
    const float logz = mx2 + fast_log(s);   // valid on every lane of the 8-group

    // ---- Gold path score for this row (8 lanes split the L axis) ----
    {
        const int b = b0 + row;
        const int* tg = tags + (size_t)b * CL;
        const unsigned char* mr = msk + (size_t)b * CL;
        const float* emr = em + (size_t)b * CL * CT;

        float em_sum = 0.0f, tr_sum = 0.0f;
        int cnt = 0;
        const int t_beg = g * (CL / 8);
        const int t_end = t_beg + (CL / 8);
        for (int t2 = t_beg; t2 < t_end; ++t2) {
            const int tag = tg[t2];
            const float mf = mr[t2] ? 1.0f : 0.0f;
            cnt += mr[t2] ? 1 : 0;
            if (t2 < CL - 1) {
                em_sum += emr[(size_t)t2 * CT + tag] * mf;
                const float mf1 = mr[t2 + 1] ? 1.0f : 0.0f;
                tr_sum += trans[tag * CT + tg[t2 + 1]] * mf1;
            }
        }
        em_sum += __shfl_xor(em_sum, 1, 32);
        em_sum += __shfl_xor(em_sum, 2, 32);
        em_sum += __shfl_xor(em_sum, 4, 32);
        tr_sum += __shfl_xor(tr_sum, 1, 32);
        tr_sum += __shfl_xor(tr_sum, 2, 32);
        tr_sum += __shfl_xor(tr_sum, 4, 32);
        cnt += __shfl_xor(cnt, 1, 32);
        cnt += __shfl_xor(cnt, 2, 32);
        cnt += __shfl_xor(cnt, 4, 32);

        if (g == 0) {
            float score = startT[tg[0]] + em_sum + tr_sum;
            const int last = cnt - 1;
            const int lt = tg[last];
            score += emr[(size_t)last * CT + lt] + endT[lt];
            // loss = mean_b(log_z - score)
            atomicAdd(out, (logz - score) * (1.0f / (float)CB));
        }
    }
}

extern "C" void kernel_launch(void* const* d_in, const int* in_sizes, int n_in,
                              void* d_out, int out_size, void* d_ws, size_t ws_size,
                              hipStream_t stream) {
    const float* em     = (const float*)d_in[0];
    const float* trans  = (const float*)d_in[1];
    const float* startT = (const float*)d_in[2];
    const float* endT   = (const float*)d_in[3];
    const int*   tags   = (const int*)d_in[4];
    const unsigned char* msk = (const unsigned char*)d_in[5];
    float* out = (float*)d_out;

    crf_zero_out<<<1, 1, 0, stream>>>(out);
    crf_forward_kernel<<<dim3(CB / 16), dim3(128), 0, stream>>>(
        em, trans, startT, endT, tags, msk, out);
}